// LocalMambaBlock_40346922778709
// MI455X (gfx1250) — compile-verified
//
#include <hip/hip_runtime.h>
#include <hip/hip_bf16.h>
#include <math.h>

typedef __attribute__((ext_vector_type(2))) float v2f;
typedef __attribute__((ext_vector_type(8))) float v8f;
typedef __attribute__((ext_vector_type(4))) int   v4i;

typedef __attribute__((address_space(1))) v4i* gv4i_p;   // global v4i*
typedef __attribute__((address_space(3))) v4i* lv4i_p;   // LDS v4i*

#define B_SZ    4
#define L_SZ    2048
#define DMODEL  1024
#define DINNER  2048
#define DCONV   4
#define DTRANK  64
#define DSTATE  16
#define NPROJ   (DTRANK + 2 * DSTATE)   /* 96 */

// ---------------------------------------------------------------------------
// CDNA5 async global->LDS copy (ASYNCcnt-tracked). Falls back to a plain
// synchronous float4 copy if the builtins are unavailable in this toolchain.
// ---------------------------------------------------------------------------
#if __has_builtin(__builtin_amdgcn_global_load_async_to_lds_b128) && \
    __has_builtin(__builtin_amdgcn_s_wait_asynccnt)
#define HAVE_ASYNC_LDS 1
#else
#define HAVE_ASYNC_LDS 0
#endif

__device__ __forceinline__ void copy_b128_to_lds(const float* __restrict__ g,
                                                 float* l)
{
#if HAVE_ASYNC_LDS
    __builtin_amdgcn_global_load_async_to_lds_b128(
        (gv4i_p)g, (lv4i_p)l, /*offset=*/0, /*cpol=*/0);
#else
    *(float4*)l = *(const float4*)g;
#endif
}

__device__ __forceinline__ void wait_async_le_4()
{
#if HAVE_ASYNC_LDS
    __builtin_amdgcn_s_wait_asynccnt(4);
#endif
}

__device__ __forceinline__ void wait_async_le_0()
{
#if HAVE_ASYNC_LDS
    __builtin_amdgcn_s_wait_asynccnt(0);
#endif
}

// ---------------------------------------------------------------------------
// Tile staging: A tile 64x16 -> As[64][20] (pad), B tile 16x64 -> Bs[16][68].
// 4 x b128 transfers per thread per chunk (2 A + 2 B). Full-N blocks only.
// ---------------------------------------------------------------------------
__device__ __forceinline__ void stage_tiles(const float* __restrict__ A, int lda,
                                            const float* __restrict__ Bm, int ldb,
                                            int m0, int n0, int k0, int tid,
                                            float* As, float* Bs)
{
    #pragma unroll
    for (int i = 0; i < 2; ++i) {
        const int s  = tid + i * 128;          // A slot 0..255
        const int r  = s >> 2;
        const int c4 = (s & 3) << 2;
        copy_b128_to_lds(A + (size_t)(m0 + r) * lda + k0 + c4, &As[r * 20 + c4]);
    }
    #pragma unroll
    for (int i = 0; i < 2; ++i) {
        const int s  = tid + i * 128;          // B slot 0..255
        const int r  = s >> 4;
        const int c4 = (s & 15) << 2;
        copy_b128_to_lds(Bm + (size_t)(k0 + r) * ldb + n0 + c4, &Bs[r * 68 + c4]);
    }
}

// ---------------------------------------------------------------------------
// Generic f32 WMMA GEMM:  C[M,N] = act(A[M,K] @ B[K,N] + bias)
// BM=64 (4 waves x 16 rows), BN=64 (4 x 16-col subtiles per wave), BK=16.
// Requires M % 64 == 0 and K % 16 == 0 (true at all call sites); N masked.
// Full-N blocks use a double-buffered async-LDS pipeline; N-edge blocks
// (only the N=96 GEMM) take a masked synchronous path.
// act: 0 = none, 1 = softplus(x + bias[col])
// ---------------------------------------------------------------------------
__global__ __launch_bounds__(128) void gemm_f32_wmma(
    const float* __restrict__ A, int lda,
    const float* __restrict__ Bm, int ldb,
    float* __restrict__ C, int ldc,
    int M, int N, int K,
    const float* __restrict__ bias, int act)
{
    __shared__ __align__(16) float As[2][64 * 20];
    __shared__ __align__(16) float Bs[2][16 * 68];

    const int tid   = threadIdx.x;
    const int lane  = tid & 31;
    const int wv    = tid >> 5;           // wave 0..3 -> M sub-strip
    const int row16 = lane & 15;
    const int kb    = (lane >> 4) << 1;   // 0 for lanes 0-15, 2 for lanes 16-31
    const int m0    = blockIdx.y * 64;
    const int n0    = blockIdx.x * 64;

    v8f acc[4];
    #pragma unroll
    for (int i = 0; i < 4; ++i)
        #pragma unroll
        for (int j = 0; j < 8; ++j)
            acc[i][j] = 0.0f;

    if (n0 + 64 <= N) {
        // ---------------- async double-buffered pipeline ----------------
        const int nchunks = K >> 4;
        stage_tiles(A, lda, Bm, ldb, m0, n0, 0, tid, As[0], Bs[0]);
        for (int i = 0; i < nchunks; ++i) {
            const int p = i & 1;
            if (i + 1 < nchunks) {
                stage_tiles(A, lda, Bm, ldb, m0, n0, (i + 1) << 4, tid,
                            As[p ^ 1], Bs[p ^ 1]);
                wait_async_le_4();     // chunk i's 4 transfers done (in-order)
            } else {
                wait_async_le_0();
            }
            __syncthreads();

            const float* Ap = As[p];
            const float* Bp = Bs[p];
            #pragma unroll
            for (int ks = 0; ks < 16; ks += 4) {
                v2f a;
                const float* ap = &Ap[(wv * 16 + row16) * 20 + ks + kb];
                a.x = ap[0];
                a.y = ap[1];
                #pragma unroll
                for (int nt = 0; nt < 4; ++nt) {
                    v2f bf;
                    bf.x = Bp[(ks + kb    ) * 68 + nt * 16 + row16];
                    bf.y = Bp[(ks + kb + 1) * 68 + nt * 16 + row16];
                    acc[nt] = __builtin_amdgcn_wmma_f32_16x16x4_f32(
                        false, a, false, bf, (short)0, acc[nt], false, false);
                }
            }
            __syncthreads();   // all waves done reading p before it is refilled
        }
    } else {
        // ---------------- masked synchronous path (N edge) ----------------
        for (int k0 = 0; k0 < K; k0 += 16) {
            for (int i = tid; i < 64 * 4; i += 128) {
                const int r  = i >> 2;
                const int c4 = (i & 3) << 2;
                *(float4*)(&As[0][r * 20 + c4]) =
                    *(const float4*)(A + (size_t)(m0 + r) * lda + k0 + c4);
            }
            for (int i = tid; i < 16 * 16; i += 128) {
                const int r   = i >> 4;
                const int c4  = (i & 15) << 2;
                const int col = n0 + c4;
                const float* src = Bm + (size_t)(k0 + r) * ldb;
                float4 v;
                v.x = (col + 0 < N) ? src[col + 0] : 0.0f;
                v.y = (col + 1 < N) ? src[col + 1] : 0.0f;
                v.z = (col + 2 < N) ? src[col + 2] : 0.0f;
                v.w = (col + 3 < N) ? src[col + 3] : 0.0f;
                *(float4*)(&Bs[0][r * 68 + c4]) = v;
            }
            __syncthreads();
            #pragma unroll
            for (int ks = 0; ks < 16; ks += 4) {
                v2f a;
                const float* ap = &As[0][(wv * 16 + row16) * 20 + ks + kb];
                a.x = ap[0];
                a.y = ap[1];
                #pragma unroll
                for (int nt = 0; nt < 4; ++nt) {
                    v2f bf;
                    bf.x = Bs[0][(ks + kb    ) * 68 + nt * 16 + row16];
                    bf.y = Bs[0][(ks + kb + 1) * 68 + nt * 16 + row16];
                    acc[nt] = __builtin_amdgcn_wmma_f32_16x16x4_f32(
                        false, a, false, bf, (short)0, acc[nt], false, false);
                }
            }
            __syncthreads();
        }
    }

    // ---- epilogue: C/D layout -> VGPR j holds rows j (lanes 0-15) / j+8 ----
    const int rbase = m0 + wv * 16 + ((lane >> 4) << 3);
    #pragma unroll
    for (int nt = 0; nt < 4; ++nt) {
        const int c = n0 + nt * 16 + row16;
        if (c < N) {
            #pragma unroll
            for (int j = 0; j < 8; ++j) {
                float v = acc[nt][j];
                if (act == 1) {
                    v += bias[c];
                    v = (v > 20.0f) ? v : log1pf(__expf(v));   // softplus
                }
                C[(size_t)(rbase + j) * ldc + c] = v;
            }
        }
    }
}

// ---------------------------------------------------------------------------
// Causal depthwise conv (width 4) + SiLU over the u half of xz.
// float4-vectorized along channels.
// ---------------------------------------------------------------------------
__global__ __launch_bounds__(256) void conv_silu_kernel(
    const float* __restrict__ xz,      // [B*L, 2*DINNER], u in cols [0,DINNER)
    const float* __restrict__ w,       // [DINNER, 4]
    const float* __restrict__ cb,      // [DINNER]
    float* __restrict__ uc)            // [B*L, DINNER]
{
    const size_t idx    = (size_t)blockIdx.x * blockDim.x + threadIdx.x;
    const size_t total4 = (size_t)B_SZ * L_SZ * (DINNER / 4);
    if (idx >= total4) return;
    const int    d4 = (int)(idx % (DINNER / 4)) * 4;
    const size_t bl = idx / (DINNER / 4);     // b*L + l
    const int    l  = (int)(bl % L_SZ);

    float wv[4][4];
    #pragma unroll
    for (int j = 0; j < 4; ++j)
        *(float4*)wv[j] = *(const float4*)(w + (size_t)(d4 + j) * DCONV);

    float accv[4];
    {
        const float4 c0 = *(const float4*)(cb + d4);
        accv[0] = c0.x; accv[1] = c0.y; accv[2] = c0.z; accv[3] = c0.w;
    }

    #pragma unroll
    for (int k = 0; k < DCONV; ++k) {
        const int ls = l - (DCONV - 1) + k;
        if (ls >= 0) {
            const float4 uv =
                *(const float4*)(xz + (bl + (size_t)(ls - l)) * (2 * DINNER) + d4);
            const float* up = (const float*)&uv;
            #pragma unroll
            for (int j = 0; j < 4; ++j)
                accv[j] += up[j] * wv[j][k];
        }
    }

    float4 o;
    o.x = accv[0] / (1.0f + __expf(-accv[0]));
    o.y = accv[1] / (1.0f + __expf(-accv[1]));
    o.z = accv[2] / (1.0f + __expf(-accv[2]));
    o.w = accv[3] / (1.0f + __expf(-accv[3]));
    *(float4*)(uc + bl * DINNER + d4) = o;
}

// ---------------------------------------------------------------------------
// Selective scan: one lane per channel, h[16] in VGPRs. B_t/C_t addresses are
// block-uniform, so they are loaded directly (scalar broadcast path) — no LDS,
// no barriers in the 2048-step sequential loop.
// Reads u from uc and writes y*silu(z) back over uc in place.
// delta lives in the (dead) u columns of xz; z in the upper columns.
// ---------------------------------------------------------------------------
__global__ __launch_bounds__(256) void scan_kernel(
    float* __restrict__ uc,
    const float* __restrict__ xz,
    const float* __restrict__ proj,    // [B*L, 96]: dt_r | Bm | Cm
    const float* __restrict__ A_log,   // [DINNER, 16]
    const float* __restrict__ Dp)      // [DINNER]
{
    const int b = blockIdx.y;
    const int d = blockIdx.x * blockDim.x + threadIdx.x;

    float Ac[DSTATE], h[DSTATE];
    #pragma unroll
    for (int n = 0; n < DSTATE; ++n) {
        Ac[n] = -__expf(A_log[d * DSTATE + n]);   // A = -exp(A_log)
        h[n]  = 0.0f;
    }
    const float dpd = Dp[d];

    for (int t = 0; t < L_SZ; ++t) {
        const size_t row = (size_t)b * L_SZ + t;
        const float* __restrict__ pr = proj + row * NPROJ + DTRANK;  // uniform

        const float u_t = uc[row * DINNER + d];
        const float dlt = xz[row * (2 * DINNER) + d];
        const float z_t = xz[row * (2 * DINNER) + DINNER + d];
        const float du  = dlt * u_t;

        float y = 0.0f;
        #pragma unroll
        for (int n = 0; n < DSTATE; ++n) {
            h[n] = h[n] * __expf(dlt * Ac[n]) + du * pr[n];
            y   += h[n] * pr[DSTATE + n];
        }
        y += u_t * dpd;
        const float sz = z_t / (1.0f + __expf(-z_t));
        uc[row * DINNER + d] = y * sz;
    }
}

// ---------------------------------------------------------------------------
extern "C" void kernel_launch(void* const* d_in, const int* in_sizes, int n_in,
                              void* d_out, int out_size, void* d_ws, size_t ws_size,
                              hipStream_t stream)
{
    (void)in_sizes; (void)n_in; (void)out_size; (void)ws_size;

    const float* x       = (const float*)d_in[0];
    const float* W_in    = (const float*)d_in[1];
    const float* conv_w  = (const float*)d_in[2];
    const float* conv_b  = (const float*)d_in[3];
    const float* W_xproj = (const float*)d_in[4];
    const float* W_dt    = (const float*)d_in[5];
    const float* b_dt    = (const float*)d_in[6];
    const float* A_log   = (const float*)d_in[7];
    const float* Dp      = (const float*)d_in[8];
    const float* W_out   = (const float*)d_in[9];
    float*       out     = (float*)d_out;

    const int Mrows = B_SZ * L_SZ;                      // 8192
    float* xz   = (float*)d_ws;                         // [8192, 4096]  134 MB
    float* uc   = xz + (size_t)Mrows * (2 * DINNER);    // [8192, 2048]   67 MB
    float* proj = uc + (size_t)Mrows * DINNER;          // [8192,   96]    3 MB

    const dim3 blk(128);

    // 1) xz = x @ W_in
    gemm_f32_wmma<<<dim3((2 * DINNER) / 64, Mrows / 64), blk, 0, stream>>>(
        x, DMODEL, W_in, 2 * DINNER, xz, 2 * DINNER,
        Mrows, 2 * DINNER, DMODEL, nullptr, 0);

    // 2) uc = silu(causal_dwconv(u))
    {
        const size_t total4 = (size_t)Mrows * (DINNER / 4);
        conv_silu_kernel<<<dim3((unsigned)((total4 + 255) / 256)), dim3(256), 0, stream>>>(
            xz, conv_w, conv_b, uc);
    }

    // 3) proj = uc @ W_xproj  (N = 96, masked tiles)
    gemm_f32_wmma<<<dim3((NPROJ + 63) / 64, Mrows / 64), blk, 0, stream>>>(
        uc, DINNER, W_xproj, NPROJ, proj, NPROJ,
        Mrows, NPROJ, DINNER, nullptr, 0);

    // 4) delta = softplus(dt_r @ W_dt + b_dt), written over dead u cols of xz
    gemm_f32_wmma<<<dim3(DINNER / 64, Mrows / 64), blk, 0, stream>>>(
        proj, NPROJ, W_dt, DINNER, xz, 2 * DINNER,
        Mrows, DINNER, DTRANK, b_dt, 1);

    // 5) selective scan (+ Dp skip + silu(z) gate), in place over uc
    scan_kernel<<<dim3(DINNER / 256, B_SZ), dim3(256), 0, stream>>>(
        uc, xz, proj, A_log, Dp);

    // 6) out = y @ W_out
    gemm_f32_wmma<<<dim3(DMODEL / 64, Mrows / 64), blk, 0, stream>>>(
        uc, DINNER, W_out, DMODEL, out, DMODEL,
        Mrows, DMODEL, DINNER, nullptr, 0);
}